// VVRQ_62070867362447
// MI455X (gfx1250) — compile-verified
//
#include <hip/hip_runtime.h>

// MI455X / gfx1250, wave32. One wave = one sample's 16x16 complex state,
// held entirely in VGPRs in V_WMMA_F32_16X16X4_F32 C/D layout.
// Gates -> fused 16x16 complex matmuls via v_wmma_f32_16x16x4_f32.

typedef __attribute__((ext_vector_type(2))) float v2f;
typedef __attribute__((ext_vector_type(8))) float v8f;

#define NQ 8
#define NLAYERS 4

struct cpx { float re, im; };

__device__ __forceinline__ cpx cmul(cpx a, cpx b) {
  cpx r; r.re = a.re * b.re - a.im * b.im; r.im = a.re * b.im + a.im * b.re; return r;
}

// Entry (i,j) of U = RZ(g) * RY(b) * RX(a)
__device__ __forceinline__ cpx u_entry(float a, float bb, float g, int i, int j) {
  float ca = cosf(0.5f * a),  sa = sinf(0.5f * a);
  float cb = cosf(0.5f * bb), sb = sinf(0.5f * bb);
  cpx m;
  if (i == 0 && j == 0) { m.re =  cb * ca; m.im =  sb * sa; }
  else if (i == 0)      { m.re = -sb * ca; m.im = -cb * sa; }
  else if (j == 0)      { m.re =  sb * ca; m.im = -cb * sa; }
  else                  { m.re =  cb * ca; m.im = -sb * sa; }
  float cg = cosf(0.5f * g), sg = sinf(0.5f * g);
  cpx ph; ph.re = cg; ph.im = (i == 0) ? -sg : sg;
  return cmul(ph, m);
}

// CNOT chain permutation: control bit3->target bit2, then b2->b1, then b1->b0,
// composed as state map s'[r] = s[perm(r)].
__host__ __device__ constexpr int chainperm(int r) {
  r ^= ((r >> 1) & 1);        // last CNOT in chain (lowest bits)
  r ^= ((r >> 2) & 1) << 1;
  r ^= ((r >> 3) & 1) << 2;   // first CNOT in chain
  return r;
}

// ---------------- prep: build 16x16 complex gate matrices from weights -------
// ws layout (floats): [layer][mat: 0=U_row,1=W_col][comp: 0=re,1=im][256]
__global__ void qprep(const float* __restrict__ w, float* __restrict__ ws) {
  int t = threadIdx.x;
  int m = t >> 4, k = t & 15;
  for (int l = 0; l < NLAYERS; ++l) {
    // U_row: wires 4..7 rotations (wire 4 <-> row bit3 ... wire 7 <-> row bit0)
    cpx pr; pr.re = 1.f; pr.im = 0.f;
    #pragma unroll
    for (int j = 3; j >= 0; --j) {
      int q = 4 + (3 - j);
      int base = (l * NQ + q) * 3;
      pr = cmul(pr, u_entry(w[base], w[base + 1], w[base + 2], (m >> j) & 1, (k >> j) & 1));
    }
    float* dst = ws + (size_t)((l * 2 + 0) * 2) * 256;
    dst[m * 16 + k]       = pr.re;
    dst[256 + m * 16 + k] = pr.im;

    // W_col: wires 0..3 rotations with column CNOT chain folded (row-permuted kron)
    int mp = chainperm(m);
    cpx pc; pc.re = 1.f; pc.im = 0.f;
    #pragma unroll
    for (int j = 3; j >= 0; --j) {
      int q = (3 - j);
      int base = (l * NQ + q) * 3;
      pc = cmul(pc, u_entry(w[base], w[base + 1], w[base + 2], (mp >> j) & 1, (k >> j) & 1));
    }
    dst = ws + (size_t)((l * 2 + 1) * 2) * 256;
    dst[m * 16 + k]       = pc.re;
    dst[256 + m * 16 + k] = pc.im;
  }
}

// ------------- layout helpers (compile-time unrolled, wave32) ----------------

// C/D layout -> B layout (same matrix, row index m remapped reg/half).
__device__ __forceinline__ void cd_to_b(const float* S, float* Bv, int half) {
  const int r0t[8] = {0, 1, 4, 5, 0, 1, 4, 5};
  #pragma unroll
  for (int d = 0; d < 8; ++d) {
    int r0 = r0t[d];
    if (d < 4) {
      float t = __shfl_xor(S[r0 + 2], 16, 32);
      Bv[d] = half ? t : S[r0];
    } else {
      float t = __shfl_xor(S[r0], 16, 32);
      Bv[d] = half ? S[r0 + 2] : t;
    }
  }
}

// Complex 16x16 matmul: (Xo + iYo) = (Are + iAim) * (Xb + iYb)
// mat: ws pointer, A-layout loaded per-lane; Xb/Yb in B layout; out in C/D.
__device__ __forceinline__ void cmulmat(const float* __restrict__ mat, int lane,
                                        const float* Xb, const float* Yb,
                                        float* Xo, float* Yo) {
  const int half = lane >> 4;
  const int mrow = lane & 15;
  float Are[8], Aim[8];
  #pragma unroll
  for (int a = 0; a < 8; ++a) {
    int k = 4 * (a >> 1) + (a & 1) + 2 * half;   // A layout: (m=lane&15, k)
    Are[a] = mat[mrow * 16 + k];
    Aim[a] = mat[256 + mrow * 16 + k];
  }
  v8f accX = {}; v8f accY = {};
  #pragma unroll
  for (int kc = 0; kc < 4; ++kc) {
    int a0 = 2 * kc, a1 = a0 + 1;
    v2f ar;  ar[0]  = Are[a0];  ar[1]  = Are[a1];
    v2f ai;  ai[0]  = Aim[a0];  ai[1]  = Aim[a1];
    v2f bx;  bx[0]  = Xb[a0];   bx[1]  = Xb[a1];
    v2f by;  by[0]  = Yb[a0];   by[1]  = Yb[a1];
    v2f nby; nby[0] = -Yb[a0];  nby[1] = -Yb[a1];
    accX = __builtin_amdgcn_wmma_f32_16x16x4_f32(false, ar, false, bx,  (short)0, accX, false, false);
    accX = __builtin_amdgcn_wmma_f32_16x16x4_f32(false, ai, false, nby, (short)0, accX, false, false);
    accY = __builtin_amdgcn_wmma_f32_16x16x4_f32(false, ar, false, by,  (short)0, accY, false, false);
    accY = __builtin_amdgcn_wmma_f32_16x16x4_f32(false, ai, false, bx,  (short)0, accY, false, false);
  }
  #pragma unroll
  for (int v = 0; v < 8; ++v) { Xo[v] = accX[v]; Yo[v] = accY[v]; }
}

// ---------------------------- main circuit kernel ----------------------------
__global__ __launch_bounds__(256) void qsim(const float* __restrict__ x,
                                            const float* __restrict__ ws,
                                            float* __restrict__ out, int B) {
  __shared__ float lds[8 * 512];     // 2KB per wave (X:256, Y:256 floats)
  const int lane  = threadIdx.x & 31;
  const int wslot = threadIdx.x >> 5;
  const int b     = blockIdx.x * 8 + wslot;
  if (b >= B) return;                // grid is exact; never taken
  const int half = lane >> 4;
  const int col  = lane & 15;
  float* myLds = lds + wslot * 512;

  // ---- angle encoding: state = prod_q RY(x_q)|0> (separable, real) ----
  float cq[8], sq[8];
  #pragma unroll
  for (int q = 0; q < 8; ++q) {
    float xv = x[b * 8 + q];
    cq[q] = cosf(0.5f * xv); sq[q] = sinf(0.5f * xv);
  }
  float colf = 1.f;
  #pragma unroll
  for (int q = 0; q < 4; ++q) colf *= ((col >> (3 - q)) & 1) ? sq[q] : cq[q];
  float X[8], Y[8];
  #pragma unroll
  for (int v = 0; v < 8; ++v) {
    int r = v + 8 * half;
    float rowf = 1.f;
    #pragma unroll
    for (int q = 4; q < 8; ++q) rowf *= ((r >> (7 - q)) & 1) ? sq[q] : cq[q];
    X[v] = colf * rowf; Y[v] = 0.f;
  }

  // ---- layers ----
  for (int l = 0; l < NLAYERS; ++l) {
    const int flip = l & 1;  // 0: state normal (rows=r); 1: state transposed (rows=c)
    const float* matA = ws + (size_t)((l * 2 + flip)       * 2) * 256;
    const float* matB = ws + (size_t)((l * 2 + (1 - flip)) * 2) * 256;

    // mult 1 on current leading index (cheap shuffle C/D->B conversion)
    float Xb[8], Yb[8];
    cd_to_b(X, Xb, half);
    cd_to_b(Y, Yb, half);
    cmulmat(matA, lane, Xb, Yb, X, Y);

    // LDS transpose: write C/D, read back as B-layout of the transpose
    #pragma unroll
    for (int v = 0; v < 8; ++v) {
      myLds[(v + 8 * half) * 16 + col]       = X[v];
      myLds[256 + (v + 8 * half) * 16 + col] = Y[v];
    }
    __syncthreads();
    #pragma unroll
    for (int bb = 0; bb < 8; ++bb) {
      int kk = 4 * (bb >> 1) + (bb & 1) + 2 * half;
      Xb[bb] = myLds[col * 16 + kk];
      Yb[bb] = myLds[256 + col * 16 + kk];
    }
    __syncthreads();

    // mult 2 on the other index; state orientation is now toggled
    cmulmat(matB, lane, Xb, Yb, X, Y);

    if (flip) {
      // state normal: crossing CNOT(3,4): if c bit0 (lane&1) flip r bit3 (half)
      #pragma unroll
      for (int v = 0; v < 8; ++v) {
        float tx = __shfl_xor(X[v], 16, 32);
        float ty = __shfl_xor(Y[v], 16, 32);
        if (lane & 1) { X[v] = tx; Y[v] = ty; }
      }
      // row CNOT chain (compile-time reg/half permutation)
      float nX[8], nY[8];
      #pragma unroll
      for (int v = 0; v < 8; ++v) {
        const int r0 = chainperm(v), r1 = chainperm(v + 8);
        float t0x = (r0 < 8)  ? X[r0 & 7] : __shfl_xor(X[r0 & 7], 16, 32);
        float t0y = (r0 < 8)  ? Y[r0 & 7] : __shfl_xor(Y[r0 & 7], 16, 32);
        float t1x = (r1 >= 8) ? X[r1 & 7] : __shfl_xor(X[r1 & 7], 16, 32);
        float t1y = (r1 >= 8) ? Y[r1 & 7] : __shfl_xor(Y[r1 & 7], 16, 32);
        nX[v] = half ? t1x : t0x;
        nY[v] = half ? t1y : t0y;
      }
      #pragma unroll
      for (int v = 0; v < 8; ++v) { X[v] = nX[v]; Y[v] = nY[v]; }
    } else {
      // state transposed: crossing CNOT(3,4): odd regs (c odd) flip r bit3 (lane bit3)
      #pragma unroll
      for (int v = 1; v < 8; v += 2) {
        X[v] = __shfl_xor(X[v], 8, 32);
        Y[v] = __shfl_xor(Y[v], 8, 32);
      }
      // row CNOT chain: rows live in lanes -> one bpermute per register
      int src = chainperm(lane & 15) | (lane & 16);
      #pragma unroll
      for (int v = 0; v < 8; ++v) {
        X[v] = __shfl(X[v], src, 32);
        Y[v] = __shfl(Y[v], src, 32);
      }
    }
  }
  // state ends in normal orientation (rows=r in regs, cols=c in lanes)

  // ---- <Z_q> = sum_i |s_i|^2 * (1 - 2*bit_q(i)) ----
  float psum = 0.f, s5 = 0.f, s6 = 0.f, s7 = 0.f;
  #pragma unroll
  for (int v = 0; v < 8; ++v) {
    float p = X[v] * X[v] + Y[v] * Y[v];
    psum += p;
    s5 += ((v >> 2) & 1) ? -p : p;   // wire5 <-> r bit2 = v bit2
    s6 += ((v >> 1) & 1) ? -p : p;   // wire6 <-> r bit1
    s7 += ((v) & 1)      ? -p : p;   // wire7 <-> r bit0
  }
  float o[8];
  o[0] = ((col >> 3) & 1) ? -psum : psum;  // wire0 <-> c bit3
  o[1] = ((col >> 2) & 1) ? -psum : psum;
  o[2] = ((col >> 1) & 1) ? -psum : psum;
  o[3] = ((col) & 1)      ? -psum : psum;
  o[4] = half ? -psum : psum;              // wire4 <-> r bit3 = half
  o[5] = s5; o[6] = s6; o[7] = s7;
  #pragma unroll
  for (int q = 0; q < 8; ++q) {
    float v = o[q];
    #pragma unroll
    for (int off = 16; off > 0; off >>= 1) v += __shfl_xor(v, off, 32);
    o[q] = v;
  }
  if (lane == 0) {
    #pragma unroll
    for (int q = 0; q < 8; ++q) out[b * 8 + q] = o[q];
  }
}

extern "C" void kernel_launch(void* const* d_in, const int* in_sizes, int n_in,
                              void* d_out, int out_size, void* d_ws, size_t ws_size,
                              hipStream_t stream) {
  const float* x = (const float*)d_in[0];
  const float* w = (const float*)d_in[1];
  float* out = (float*)d_out;
  float* ws  = (float*)d_ws;    // needs 16 KB: 4 layers x 2 mats x 2 comps x 256 f32
  int B = in_sizes[0] / NQ;

  qprep<<<1, 256, 0, stream>>>(w, ws);
  int nblocks = (B + 7) / 8;    // 8 waves (samples) per 256-thread block
  qsim<<<nblocks, 256, 0, stream>>>(x, ws, out, B);
}